// HarmonicConvolutionFilter_50388556316776
// MI455X (gfx1250) — compile-verified
//
#include <hip/hip_runtime.h>
#include <hip/hip_bf16.h>

typedef float v2f __attribute__((ext_vector_type(2)));
typedef float v4f __attribute__((ext_vector_type(4)));
typedef float v8f __attribute__((ext_vector_type(8)));
typedef unsigned int u32;
typedef u32 u32x4 __attribute__((ext_vector_type(4)));
typedef int  i32x4 __attribute__((ext_vector_type(4)));
typedef int  i32x8 __attribute__((ext_vector_type(8)));

#define Bc   4
#define Sc   1024
#define Fc   512
#define Cc   4
#define Thw  2          // time half-window T
#define TAPS 5          // 2T+1
#define Kc   4          // harmonics
#define Oc   4          // out filters
#define ROW  (Fc * Cc)  // 2048 floats of real data per time row (8 KB)
#define PAD  128        // zeroed tail per row: 512 B keeps row stride % 512 == 0
#define ROWP (ROW + PAD)// 2176 floats (8704 B = 17*512) -> stride64 2addr still legal

// One workgroup (8 wave32) per (b, s).
//  - TDM (tensor_load_to_lds) stages the 5 time rows into padded LDS rows
//  - out-of-band harmonic gathers are redirected into the zeroed row pad
//    (address cndmask) => no mask multiplies, one branch-free WMMA body
//  - each wave accumulates 16-f tiles with V_WMMA_F32_16X16X4_F32, skipping
//    (t,k) steps whose whole tile is out of band via scalar branches
//  - results are transposed through per-wave LDS scratch into one
//    coalesced global_store_b128 per tile
__global__ __launch_bounds__(256)
void HarmonicConvolutionFilter_50388556316776_kernel(const float* __restrict__ x,
                                                     const float* __restrict__ W,
                                                     float* __restrict__ out)
{
    __shared__ float ldsx[TAPS * ROWP];      // ~42.5 KB staged input rows (+pads)
    __shared__ float ldso[8][16 * Oc];       // 2 KB per-wave output transpose scratch

    const int bs   = blockIdx.x;             // 0 .. B*S-1
    const int b    = bs >> 10;               // / Sc
    const int s    = bs & (Sc - 1);          // % Sc
    const int tid  = threadIdx.x;
    const int lane = tid & 31;
    // wave id is uniform within a wave; make that visible to the compiler
    const int wave = __builtin_amdgcn_readfirstlane(tid >> 5);

    // ---- zero the row pads (out-of-band gather target) ----
    {
        const v4f z = {0.f, 0.f, 0.f, 0.f};
        for (int q = tid; q < TAPS * (PAD / 4); q += 256) {
            const int t = q / (PAD / 4);
            const int j = q % (PAD / 4);
            *(v4f*)(ldsx + (size_t)t * ROWP + ROW + (size_t)j * 4) = z;
        }
    }

    // ---- zero-fill rows that fall in the time zero-padding ----
    #pragma unroll
    for (int t = 0; t < TAPS; ++t) {
        const int tau = s + t - Thw;
        if ((unsigned)tau >= (unsigned)Sc) {           // scalar branch (blockIdx-derived)
            for (int q = tid; q < ROW / 4; q += 256) {
                v4f z = {0.f, 0.f, 0.f, 0.f};
                *(v4f*)(ldsx + (size_t)t * ROWP + (size_t)q * 4) = z;
            }
        }
    }

    // ---- TDM: wave 0 issues one tensor_load_to_lds per valid time row ----
    if (wave == 0) {
        const u32 lds0 = (u32)(size_t)(&ldsx[0]);      // LDS byte offset (flat addr low 32b)
        #pragma unroll
        for (int t = 0; t < TAPS; ++t) {
            const int tau = s + t - Thw;
            if ((unsigned)tau < (unsigned)Sc) {
                const unsigned long long ga =
                    (unsigned long long)(size_t)(x + (((size_t)b * Sc + tau) * Fc) * Cc);
                // D# group 0: count=1 | lds_addr | global_addr[56:0] | type=2
                u32x4 g0;
                g0[0] = 1u;
                g0[1] = lds0 + (u32)(t * ROWP * 4);
                g0[2] = (u32)(ga & 0xffffffffu);
                g0[3] = (u32)((ga >> 32) & 0x01ffffffu) | (2u << 30);
                // D# group 1: data_size=4B, tensor 2048x1, tile 2048x1, strides 2048
                i32x8 g1;
                g1[0] = (int)(2u << 16);           // workgroup_mask=0, data_size=2 (4B)
                g1[1] = (int)((u32)ROW << 16);     // tensor_dim0[15:0]=2048 (atomic_bar=0)
                g1[2] = (int)(1u << 16);           // tensor_dim0[31:16]=0, tensor_dim1=1
                g1[3] = (int)((u32)ROW << 16);     // tile_dim0=2048
                g1[4] = 1;                         // tile_dim1=1, tile_dim2=0
                g1[5] = ROW;                       // tensor_dim0_stride[31:0]=2048
                g1[6] = (int)((u32)ROW << 16);     // stride0[47:32]=0, stride1[15:0]=2048
                g1[7] = 0;                         // stride1[47:16]=0
                i32x4 z4 = {0, 0, 0, 0};           // groups 2/3 unused (<=2D tensor)
                i32x8 z8 = {0, 0, 0, 0, 0, 0, 0, 0};
                __builtin_amdgcn_tensor_load_to_lds(g0, g1, z4, z4, z8, 0);
            }
        }
        __builtin_amdgcn_s_wait_tensorcnt(0);
    }
    __syncthreads();

    // ---- precompute B operands: W[t,k,:,:] as 4x16 f32 tile, cols >= O are zero.
    // B layout (4x16, 2 VGPRs): lane l holds B[kB][n], n=l%16,
    //   V0: kB=(l<16)?0:2, V1: kB+1 (kB indexes C, the contraction dim).
    const int n  = lane & 15;
    const int c0 = (lane < 16) ? 0 : 2;
    v2f Bop[TAPS][Kc];
    #pragma unroll
    for (int t = 0; t < TAPS; ++t) {
        #pragma unroll
        for (int k = 0; k < Kc; ++k) {
            float bx = 0.f, by = 0.f;
            if (n < Oc) {
                bx = W[((t * Kc + k) * Cc + c0)     * Oc + n];
                by = W[((t * Kc + k) * Cc + c0 + 1) * Oc + n];
            }
            Bop[t][k].x = bx;
            Bop[t][k].y = by;
        }
    }

    // ---- each wave: f-tiles of 16; tile/fbase are SGPR-uniform -> s_cbranch skips ----
    const int m = lane & 15;                 // A row (f within tile)
    for (int tile = wave; tile < Fc / 16; tile += 8) {
        const int fbase = tile * 16;         // uniform (SGPR)
        v8f acc = {};
        #pragma unroll
        for (int k = 0; k < Kc; ++k) {
            if (fbase * (k + 1) < Fc) {      // scalar: tile touches harmonic band
                const int h   = (fbase + m) * (k + 1);
                // out-of-band lanes read zeros from the row pad (no mask multiply)
                const int off = ((h < Fc) ? h * Cc : ROW) + c0;
                #pragma unroll
                for (int t = 0; t < TAPS; ++t) {
                    const float* p = ldsx + (size_t)t * ROWP + off;
                    v2f a;
                    a.x = p[0];
                    a.y = p[1];
                    acc = __builtin_amdgcn_wmma_f32_16x16x4_f32(
                            /*neg_a=*/false, a,
                            /*neg_b=*/false, Bop[t][k],
                            /*c_mod=*/(short)0, acc,
                            /*reuse_a=*/false, /*reuse_b=*/false);
                }
            }
        }
        // ---- transpose D tile through per-wave LDS, then one coalesced B128 store ----
        // D (16x16 f32): lanes 0-15 -> M=j, N=lane; lanes 16-31 -> M=8+j, N=lane-16
        if (n < Oc) {
            #pragma unroll
            for (int j = 0; j < 8; ++j) {
                const int mm = (lane < 16) ? j : (8 + j);
                ldso[wave][mm * Oc + n] = acc[j];
            }
        }
        asm volatile("s_wait_dscnt 0" ::: "memory");   // same-wave DS RAW ordering
        if (lane < 16) {
            v4f r = *(const v4f*)&ldso[wave][lane * Oc];
            *(v4f*)(out + (((size_t)b * Sc + s) * Fc + fbase + lane) * Oc) = r;
        }
    }
}

extern "C" void kernel_launch(void* const* d_in, const int* in_sizes, int n_in,
                              void* d_out, int out_size, void* d_ws, size_t ws_size,
                              hipStream_t stream) {
    const float* x  = (const float*)d_in[0];   // [B,S,F,C] f32
    const float* W  = (const float*)d_in[1];   // [5,K,C,O] f32
    float* out      = (float*)d_out;           // [B,S,F,O] f32

    dim3 grid(Bc * Sc);                        // 4096 workgroups, one per (b,s)
    dim3 block(256);                           // 8 wave32
    hipLaunchKernelGGL(HarmonicConvolutionFilter_50388556316776_kernel,
                       grid, block, 0, stream, x, W, out);
}